// PhotometricLoss_89824946029311
// MI455X (gfx1250) — compile-verified
//
#include <hip/hip_runtime.h>
#include <math.h>

#define HW   256
#define BOX  16
#define NPIX (HW * HW)

typedef __attribute__((ext_vector_type(2))) float v2f;
typedef __attribute__((ext_vector_type(8))) float v8f;

// ---------------------------------------------------------------------------
// 1) Per-intra-box-position median over the 256 boxes (stable order statistic
//    index 127 == torch lower median for n=256). One block per position.
// ---------------------------------------------------------------------------
__global__ void median_kernel(const float* __restrict__ gt, float* __restrict__ med) {
    __shared__ float vals[256];
    const int p = blockIdx.x;          // intra-box position: p = i*16 + j
    const int i = p >> 4, j = p & 15;
    const int m = threadIdx.x;         // box index: bh*16 + bw
    const int bh = m >> 4, bw = m & 15;
    const float x = gt[(bh * BOX + i) * HW + (bw * BOX + j)];
    vals[m] = x;
    __syncthreads();
    int cnt = 0;
    for (int k = 0; k < 256; ++k) {
        const float y = vals[k];
        cnt += (y < x) || (y == x && k < m);   // stable rank
    }
    if (cnt == 127) med[p] = x;               // exactly one thread matches
}

// Tridiagonal band matrix S: S[r,c] = 1 if |r-c| <= 1 else 0 (3-tap box).
__device__ __forceinline__ float sband(int r, int c) {
    const int d = r - c;
    return (d >= -1 && d <= 1) ? 1.0f : 0.0f;
}

// ---------------------------------------------------------------------------
// 2) Row blur via WMMA:  T = Bkg * S   (Bkg[y,x] = med[y>>4, x>>4])
//    One wave (32 threads) per 16x16 output tile; band limits K to 6 chunks.
// ---------------------------------------------------------------------------
__global__ void blur_row_wmma(const float* __restrict__ med, float* __restrict__ T) {
    const int J = blockIdx.x, I = blockIdx.y;
    const int lane = threadIdx.x;
    const int half = lane >> 4;        // K sub-group select
    const int n    = lane & 15;        // M for A, N for B/D
    const int row0 = I * 16, col0 = J * 16;
    const int mrow = (row0 + n) >> 4;  // background is tile-constant per 16 rows

    v8f c = {};
    int c0 = 4 * J - 1; if (c0 < 0)  c0 = 0;
    int c1 = 4 * J + 4; if (c1 > 63) c1 = 63;
    for (int ch = c0; ch <= c1; ++ch) {
        const int kb = ch * 4 + 2 * half;
        v2f a, b;
        // A[m, k] = Bkg[row0+m, k] = med[(row0+m)>>4, k>>4]
        a[0] = med[mrow * 16 + ((kb + 0) >> 4)];
        a[1] = med[mrow * 16 + ((kb + 1) >> 4)];
        // B[k, n] = S[k, col0+n]
        b[0] = sband(kb + 0, col0 + n);
        b[1] = sband(kb + 1, col0 + n);
        c = __builtin_amdgcn_wmma_f32_16x16x4_f32(false, a, false, b,
                                                  (short)0, c, false, false);
    }
    for (int r = 0; r < 8; ++r)
        T[(row0 + r + 8 * half) * HW + col0 + n] = c[r];
}

// ---------------------------------------------------------------------------
// 3) Column blur via WMMA:  Bk = (S * T) / 9
// ---------------------------------------------------------------------------
__global__ void blur_col_wmma(const float* __restrict__ T, float* __restrict__ Bk) {
    const int J = blockIdx.x, I = blockIdx.y;
    const int lane = threadIdx.x;
    const int half = lane >> 4;
    const int n    = lane & 15;
    const int row0 = I * 16, col0 = J * 16;

    v8f c = {};
    int c0 = 4 * I - 1; if (c0 < 0)  c0 = 0;
    int c1 = 4 * I + 4; if (c1 > 63) c1 = 63;
    for (int ch = c0; ch <= c1; ++ch) {
        const int kb = ch * 4 + 2 * half;
        v2f a, b;
        // A[m, k] = S[row0+m, k]
        a[0] = sband(row0 + n, kb + 0);
        a[1] = sband(row0 + n, kb + 1);
        // B[k, n] = T[k, col0+n]
        b[0] = T[(kb + 0) * HW + col0 + n];
        b[1] = T[(kb + 1) * HW + col0 + n];
        c = __builtin_amdgcn_wmma_f32_16x16x4_f32(false, a, false, b,
                                                  (short)0, c, false, false);
    }
    const float inv9 = 1.0f / 9.0f;   // avg_pool2d count_include_pad=True
    for (int r = 0; r < 8; ++r)
        Bk[(row0 + r + 8 * half) * HW + col0 + n] = c[r] * inv9;
}

// ---------------------------------------------------------------------------
// 4) data = gt - bkg ; pred_b = pred - bkg
// ---------------------------------------------------------------------------
__global__ void sub_kernel(const float* __restrict__ gt, const float* __restrict__ pred,
                           const float* __restrict__ bk,
                           float* __restrict__ data, float* __restrict__ predb) {
    const int idx = blockIdx.x * blockDim.x + threadIdx.x;
    if (idx < NPIX) {
        const float b = bk[idx];
        data[idx]  = gt[idx]   - b;
        predb[idx] = pred[idx] - b;
    }
}

// ---------------------------------------------------------------------------
// 5) Sigma-clipped std: 6 serially-dependent full-image reductions done in a
//    single persistent workgroup (1024 threads, 64 elems/thread, LDS tree).
// ---------------------------------------------------------------------------
__global__ void sigma_clip_kernel(const float* __restrict__ data, float* __restrict__ stats) {
    __shared__ float s_sum[1024], s_ssq[1024], s_cnt[1024];
    __shared__ float sMean, sStd;
    const int tid = threadIdx.x;
    float mean = 0.0f, std = 0.0f;

    for (int pass = 0; pass < 6; ++pass) {
        float sum = 0.0f, ssq = 0.0f, cnt = 0.0f;
        for (int i = tid; i < NPIX; i += 1024) {
            const float d = data[i];
            const bool ok = (pass == 0) || (fabsf(d - mean) <= 3.0f * std);
            if (ok) { sum += d; ssq += d * d; cnt += 1.0f; }
        }
        s_sum[tid] = sum; s_ssq[tid] = ssq; s_cnt[tid] = cnt;
        __syncthreads();
        for (int off = 512; off > 0; off >>= 1) {
            if (tid < off) {
                s_sum[tid] += s_sum[tid + off];
                s_ssq[tid] += s_ssq[tid + off];
                s_cnt[tid] += s_cnt[tid + off];
            }
            __syncthreads();
        }
        if (tid == 0) {
            const float C  = s_cnt[0];
            const float mu = s_sum[0] / C;
            const float var = (s_ssq[0] - C * mu * mu) / (C - 1.0f);  // unbiased
            sMean = mu;
            sStd  = sqrtf(var);
        }
        __syncthreads();
        mean = sMean; std = sStd;
        __syncthreads();
    }
    if (tid == 0) stats[0] = std;
}

// ---------------------------------------------------------------------------
// 6) Peaks: (data > 11*std) && (data == 3x3 local max)
// ---------------------------------------------------------------------------
__global__ void peak_kernel(const float* __restrict__ data, const float* __restrict__ stats,
                            float* __restrict__ pf) {
    const int idx = blockIdx.x * blockDim.x + threadIdx.x;
    if (idx >= NPIX) return;
    const int y = idx >> 8, x = idx & 255;
    const float d = data[idx];
    float mx = -INFINITY;
    for (int dy = -1; dy <= 1; ++dy) {
        const int yy = y + dy; if (yy < 0 || yy >= HW) continue;
        for (int dx = -1; dx <= 1; ++dx) {
            const int xx = x + dx; if (xx < 0 || xx >= HW) continue;
            mx = fmaxf(mx, data[yy * HW + xx]);
        }
    }
    const float thr = 11.0f * stats[0];
    pf[idx] = (d > thr && d == mx) ? 1.0f : 0.0f;
}

// ---------------------------------------------------------------------------
// 7) Separable 9x9 dilation (max-reduce, init 0.0 like reduce_window)
// ---------------------------------------------------------------------------
__global__ void dilate_h(const float* __restrict__ in, float* __restrict__ out) {
    const int idx = blockIdx.x * blockDim.x + threadIdx.x;
    if (idx >= NPIX) return;
    const int y = idx >> 8, x = idx & 255;
    float mx = 0.0f;
    for (int dx = -4; dx <= 4; ++dx) {
        const int xx = x + dx; if (xx < 0 || xx >= HW) continue;
        mx = fmaxf(mx, in[y * HW + xx]);
    }
    out[idx] = mx;
}

__global__ void dilate_v(const float* __restrict__ in, float* __restrict__ out) {
    const int idx = blockIdx.x * blockDim.x + threadIdx.x;
    if (idx >= NPIX) return;
    const int y = idx >> 8, x = idx & 255;
    float mx = 0.0f;
    for (int dy = -4; dy <= 4; ++dy) {
        const int yy = y + dy; if (yy < 0 || yy >= HW) continue;
        mx = fmaxf(mx, in[yy * HW + x]);
    }
    out[idx] = mx;
}

// ---------------------------------------------------------------------------
// 8) Final masked photometry + relative error (single workgroup reduce)
// ---------------------------------------------------------------------------
__global__ void final_kernel(const float* __restrict__ data, const float* __restrict__ predb,
                             const float* __restrict__ mask, float* __restrict__ out) {
    __shared__ float s1[1024], s2[1024];
    const int tid = threadIdx.x;
    float a = 0.0f, b = 0.0f;
    for (int i = tid; i < NPIX; i += 1024) {
        const float m = mask[i];
        a += data[i]  * m;
        b += predb[i] * m;
    }
    s1[tid] = a; s2[tid] = b;
    __syncthreads();
    for (int off = 512; off > 0; off >>= 1) {
        if (tid < off) { s1[tid] += s1[tid + off]; s2[tid] += s2[tid + off]; }
        __syncthreads();
    }
    if (tid == 0) out[0] = fabsf(s1[0] - s2[0]) / fabsf(s1[0]);
}

// ---------------------------------------------------------------------------
extern "C" void kernel_launch(void* const* d_in, const int* in_sizes, int n_in,
                              void* d_out, int out_size, void* d_ws, size_t ws_size,
                              hipStream_t stream) {
    const float* pred = (const float*)d_in[0];   // setup_inputs order: pred, gt
    const float* gt   = (const float*)d_in[1];
    float* out = (float*)d_out;

    float* ws    = (float*)d_ws;
    float* med   = ws;            // 256 floats
    float* stats = ws + 256;      // few scalars
    float* B1    = ws + 1024;     // smoothed bkg -> dilate temp
    float* B2    = B1 + NPIX;     // row-blur temp -> data
    float* B3    = B2 + NPIX;     // pred_b
    float* B4    = B3 + NPIX;     // peak flags -> final mask

    median_kernel<<<256, 256, 0, stream>>>(gt, med);

    dim3 tiles(16, 16);
    blur_row_wmma<<<tiles, 32, 0, stream>>>(med, B2);      // T = Bkg * S
    blur_col_wmma<<<tiles, 32, 0, stream>>>(B2, B1);       // bkg = S * T / 9

    sub_kernel<<<NPIX / 256, 256, 0, stream>>>(gt, pred, B1, B2, B3);

    sigma_clip_kernel<<<1, 1024, 0, stream>>>(B2, stats);

    peak_kernel<<<NPIX / 256, 256, 0, stream>>>(B2, stats, B4);
    dilate_h<<<NPIX / 256, 256, 0, stream>>>(B4, B1);
    dilate_v<<<NPIX / 256, 256, 0, stream>>>(B1, B4);

    final_kernel<<<1, 1024, 0, stream>>>(B2, B3, B4, out);
}